// Prototype_81836306858007
// MI455X (gfx1250) — compile-verified
//
#include <hip/hip_runtime.h>
#include <math.h>

typedef __attribute__((ext_vector_type(2))) float v2f;
typedef __attribute__((ext_vector_type(8))) float v8f;

#define NROW   65536   // T*N_ rows of qr
#define KDIM   256     // DIMS
#define MKEYS  64      // M
#define ROWS1  128     // rows per block in score GEMM (8 waves x 16)
#define TILE4  64      // rows per block in row kernel

// ---- monotone uint encoding for float atomic max ----
__device__ __forceinline__ unsigned fenc(float f) {
  unsigned u = __float_as_uint(f);
  return (u & 0x80000000u) ? ~u : (u | 0x80000000u);
}
__device__ __forceinline__ float fdec(unsigned u) {
  u = (u & 0x80000000u) ? (u & 0x7fffffffu) : ~u;
  return __uint_as_float(u);
}
#define ENC_NEG_INF 0x007fffffu   // fenc(-inf)

// -------------------------------------------------------------------
// Kernel 0: init workspace accumulators
// -------------------------------------------------------------------
__global__ void k0_init(float* ws_qu, unsigned* ws_colmax, float* ws_colsum) {
  int t = threadIdx.x;
  for (int it = 0; it < 64; ++it) ws_qu[it * 256 + t] = 0.0f;   // 64*256 = 16384
  if (t < MKEYS) { ws_colmax[t] = ENC_NEG_INF; ws_colsum[t] = 0.0f; }
}

// -------------------------------------------------------------------
// Kernel 1: score = qr @ keys^T  via V_WMMA_F32_16X16X4_F32
//   qr[row,d] = query[d*NROW + row]  (rows contiguous -> coalesced)
//   keys (64x256 = 64KB) staged in LDS.
//   block: 256 thr = 8 waves, each wave: 16 rows x 64 cols (4 accs)
// -------------------------------------------------------------------
__global__ void k1_score(const float* __restrict__ q,
                         const float* __restrict__ keys,
                         float* __restrict__ score) {
  __shared__ float sk[MKEYS * KDIM];  // 64 KB
  int t = threadIdx.x;
  {
    const float4* k4 = (const float4*)keys;
    float4* s4 = (float4*)sk;
#pragma unroll
    for (int it = 0; it < 16; ++it) s4[it * 256 + t] = k4[it * 256 + t];
  }
  __syncthreads();

  int lid = t & 31, wid = t >> 5;
  int m = lid & 15, kh = lid >> 4;          // A lane decomposition
  size_t i0 = (size_t)blockIdx.x * ROWS1 + (size_t)wid * 16;

  v8f acc0 = {}, acc1 = {}, acc2 = {}, acc3 = {};
  for (int d0 = 0; d0 < KDIM; d0 += 4) {
    v2f a;
    a.x = q[(size_t)(d0 + 2 * kh) * NROW + i0 + m];
    a.y = q[(size_t)(d0 + 2 * kh + 1) * NROW + i0 + m];
    v2f b0, b1, b2, b3;
    b0.x = sk[(0 * 16 + m) * KDIM + d0 + 2 * kh];  b0.y = sk[(0 * 16 + m) * KDIM + d0 + 2 * kh + 1];
    b1.x = sk[(1 * 16 + m) * KDIM + d0 + 2 * kh];  b1.y = sk[(1 * 16 + m) * KDIM + d0 + 2 * kh + 1];
    b2.x = sk[(2 * 16 + m) * KDIM + d0 + 2 * kh];  b2.y = sk[(2 * 16 + m) * KDIM + d0 + 2 * kh + 1];
    b3.x = sk[(3 * 16 + m) * KDIM + d0 + 2 * kh];  b3.y = sk[(3 * 16 + m) * KDIM + d0 + 2 * kh + 1];
    acc0 = __builtin_amdgcn_wmma_f32_16x16x4_f32(false, a, false, b0, (short)0, acc0, false, false);
    acc1 = __builtin_amdgcn_wmma_f32_16x16x4_f32(false, a, false, b1, (short)0, acc1, false, false);
    acc2 = __builtin_amdgcn_wmma_f32_16x16x4_f32(false, a, false, b2, (short)0, acc2, false, false);
    acc3 = __builtin_amdgcn_wmma_f32_16x16x4_f32(false, a, false, b3, (short)0, acc3, false, false);
  }
  // C layout: VGPR j -> row j (lanes 0-15) / row j+8 (lanes 16-31), col = lane&15
#pragma unroll
  for (int j = 0; j < 8; ++j) {
    size_t row = i0 + j + (size_t)kh * 8;
    score[row * MKEYS +  0 + m] = acc0[j];
    score[row * MKEYS + 16 + m] = acc1[j];
    score[row * MKEYS + 32 + m] = acc2[j];
    score[row * MKEYS + 48 + m] = acc3[j];
  }
}

// -------------------------------------------------------------------
// Kernel 2: per-column max of score (encoded-uint atomic max)
// -------------------------------------------------------------------
__global__ void k2_colmax(const float* __restrict__ score, unsigned* ws_colmax) {
  __shared__ unsigned lmax[MKEYS];
  int t = threadIdx.x;
  if (t < MKEYS) lmax[t] = ENC_NEG_INF;
  __syncthreads();
  size_t base = (size_t)blockIdx.x * 256 * MKEYS;
  for (int it = 0; it < 64; ++it) {
    int f = it * 256 + t;
    atomicMax(&lmax[f & 63], fenc(score[base + f]));
  }
  __syncthreads();
  if (t < MKEYS) atomicMax(&ws_colmax[t], lmax[t]);
}

// -------------------------------------------------------------------
// Kernel 3: per-column sum of exp(score - colmax)
// -------------------------------------------------------------------
__global__ void k3_colsum(const float* __restrict__ score,
                          const unsigned* __restrict__ ws_colmax,
                          float* ws_colsum) {
  __shared__ float lsum[MKEYS];
  __shared__ float cmx[MKEYS];
  int t = threadIdx.x;
  if (t < MKEYS) { lsum[t] = 0.0f; cmx[t] = fdec(ws_colmax[t]); }
  __syncthreads();
  size_t base = (size_t)blockIdx.x * 256 * MKEYS;
  for (int it = 0; it < 64; ++it) {
    int f = it * 256 + t;
    int j = f & 63;
    atomicAdd(&lsum[j], __expf(score[base + f] - cmx[j]));
  }
  __syncthreads();
  if (t < MKEYS) atomicAdd(&ws_colsum[t], lsum[t]);
}

// -------------------------------------------------------------------
// Kernel 4: per-64-row tile: softmaxes, top2, losses, concat GEMM (WMMA),
//           updated_query (both halves). 256 threads, ~164KB LDS.
// -------------------------------------------------------------------
__global__ void __launch_bounds__(256, 1)
k4_rows(const float* __restrict__ q,
        const float* __restrict__ keys,
        const float* __restrict__ score,
        const unsigned* __restrict__ ws_colmax,
        const float* __restrict__ ws_colsum,
        int* __restrict__ ws_g, float* __restrict__ ws_w,
        float* __restrict__ out_uq, float* __restrict__ out_sq,
        float* __restrict__ out_smem, float* __restrict__ out_spread,
        float* __restrict__ out_gath) {
  __shared__ float sk[MKEYS * KDIM];      // keys, 64 KB
  __shared__ float qr[KDIM * 65];         // qr tile [d][r], pitch 65; reused for cm
  __shared__ float sc[TILE4 * 65];        // score tile [r][j], pitch 65
  __shared__ float sm[TILE4 * 65];        // score_memory tile
  __shared__ float cmx[MKEYS], csum[MKEYS];
  __shared__ float rmax[TILE4], rsum[TILE4], posA[TILE4], negA[TILE4];
  __shared__ int   gA[TILE4], g2A[TILE4];

  int t = threadIdx.x;
  size_t i0 = (size_t)blockIdx.x * TILE4;

  // ---- phase 1: stage keys, colmax/colsum, qr tile; emit updated_query[0:256]
  {
    const float4* k4p = (const float4*)keys;
    float4* s4 = (float4*)sk;
#pragma unroll
    for (int it = 0; it < 16; ++it) s4[it * 256 + t] = k4p[it * 256 + t];
  }
  if (t < MKEYS) { cmx[t] = fdec(ws_colmax[t]); csum[t] = ws_colsum[t]; posA[t] = 0.0f; negA[t] = 0.0f; }
  for (int it = 0; it < 64; ++it) {
    int f = it * 256 + t;
    int r = f & 63, d = f >> 6;
    float v = q[(size_t)d * NROW + i0 + r];
    qr[d * 65 + r] = v;
    out_uq[(size_t)d * NROW + i0 + r] = v;        // first half = qr copy
  }
  __syncthreads();

  // ---- phase 2: stage score tile
  for (int it = 0; it < 16; ++it) {
    int f = it * 256 + t;
    sc[(f >> 6) * 65 + (f & 63)] = score[i0 * MKEYS + f];
  }
  __syncthreads();

  // ---- phase 3: per-row stats (top2, row max, row sumexp, gather weight)
  if (t < TILE4) {
    float m1 = -3.4e38f, m2 = -3.4e38f; int i1 = 0, i2 = 0;
    for (int j = 0; j < MKEYS; ++j) {
      float s = sc[t * 65 + j];
      if (s > m1) { m2 = m1; i2 = i1; m1 = s; i1 = j; }
      else if (s > m2) { m2 = s; i2 = j; }
    }
    float sum = 0.0f;
    for (int j = 0; j < MKEYS; ++j) sum += __expf(sc[t * 65 + j] - m1);
    rmax[t] = m1; rsum[t] = sum; gA[t] = i1; g2A[t] = i2;
    ws_g[i0 + t] = i1;
    ws_w[i0 + t] = __expf(sc[t * 65 + i1] - cmx[i1]);   // score_query/colmax(sq) simplifies
  }
  __syncthreads();

  // ---- phase 4: write score_query & score_memory, fill sm tile
  for (int it = 0; it < 16; ++it) {
    int f = it * 256 + t;
    int r = f >> 6, j = f & 63;
    float s = sc[r * 65 + j];
    float smv = __expf(s - rmax[r]) / rsum[r];
    sm[r * 65 + j] = smv;
    out_smem[i0 * MKEYS + f] = smv;
    out_sq[i0 * MKEYS + f] = __expf(s - cmx[j]) / csum[j];
  }
  __syncthreads();

  // ---- phase 5: gathering loss + triplet distances (thread t = dim d)
  for (int rr = 0; rr < TILE4; ++rr) {
    int g = gA[rr], g2 = g2A[rr];
    float qv = qr[t * 65 + rr];
    float kp = sk[g * KDIM + t];
    float kn = sk[g2 * KDIM + t];
    float diff = qv - kp;
    out_gath[(i0 + rr) * (size_t)KDIM + t] = diff * diff;
    float dp = diff + 1e-6f;           float dp2 = dp * dp;
    float dn = (qv - kn) + 1e-6f;      float dn2 = dn * dn;
#pragma unroll
    for (int off = 16; off > 0; off >>= 1) {
      dp2 += __shfl_xor(dp2, off);
      dn2 += __shfl_xor(dn2, off);
    }
    if ((t & 31) == 0) { atomicAdd(&posA[rr], dp2); atomicAdd(&negA[rr], dn2); }
  }
  __syncthreads();
  if (t < TILE4) {
    float dpos = sqrtf(posA[t]), dneg = sqrtf(negA[t]);
    out_spread[i0 + t] = fmaxf(dpos - dneg + 1.0f, 0.0f);
  }
  __syncthreads();   // qr reads done; safe to overwrite with cm

  // ---- phase 6: concat_memory = sm(64x64) @ keys(64x256) via WMMA f32 16x16x4
  {
    int lid = t & 31, wid = t >> 5;
    int m = lid & 15, kh = lid >> 4;
#pragma unroll
    for (int s = 0; s < 8; ++s) {
      int tid = wid + 8 * s;             // 64 tiles: 4 row-tiles x 16 d-tiles
      int r0 = (tid & 3) * 16;
      int d0 = (tid >> 2) * 16;
      v8f acc = {};
      for (int k0 = 0; k0 < MKEYS; k0 += 4) {
        v2f a, b;
        a.x = sm[(r0 + m) * 65 + k0 + 2 * kh];
        a.y = sm[(r0 + m) * 65 + k0 + 2 * kh + 1];
        b.x = sk[(k0 + 2 * kh) * KDIM + d0 + m];
        b.y = sk[(k0 + 2 * kh + 1) * KDIM + d0 + m];
        acc = __builtin_amdgcn_wmma_f32_16x16x4_f32(false, a, false, b, (short)0, acc, false, false);
      }
#pragma unroll
      for (int j = 0; j < 8; ++j)
        qr[(d0 + m) * 65 + r0 + kh * 8 + j] = acc[j];   // cm stored [d][r]
    }
  }
  __syncthreads();

  // ---- phase 7: write updated_query[256:512] (coalesced)
  for (int it = 0; it < 64; ++it) {
    int f = it * 256 + t;
    int r = f & 63, d = f >> 6;
    out_uq[(size_t)(KDIM + d) * NROW + i0 + r] = qr[d * 65 + r];
  }
}

// -------------------------------------------------------------------
// Kernel 5: query_update scatter: LDS 64x256 accumulator (pitch 257),
//   1024 rows per block, then one dense atomic flush.
// -------------------------------------------------------------------
__global__ void __launch_bounds__(256, 1)
k5_scatter(const float* __restrict__ q,
           const int* __restrict__ ws_g, const float* __restrict__ ws_w,
           float* __restrict__ ws_qu) {
  __shared__ float acc[MKEYS * 257];
  int t = threadIdx.x;
  for (int it = 0; it < 65; ++it) {
    int f = it * 256 + t;
    if (f < MKEYS * 257) acc[f] = 0.0f;
  }
  __syncthreads();
  size_t i0 = (size_t)blockIdx.x * 1024;
  for (int c = 0; c < 4; ++c) {
    size_t row = i0 + c * 256 + t;
    int g = ws_g[row];
    float w = ws_w[row];
    for (int d = 0; d < KDIM; ++d) {
      float v = q[(size_t)d * NROW + row];     // coalesced across threads
      atomicAdd(&acc[g * 257 + d], w * v);     // pitch-257: conflict-free banks
    }
  }
  __syncthreads();
  for (int it = 0; it < 64; ++it) {
    int f = it * 256 + t;                      // 16384 elems
    atomicAdd(&ws_qu[f], acc[(f >> 8) * 257 + (f & 255)]);
  }
}

// -------------------------------------------------------------------
// Kernel 6: updated_memory = l2_normalize(query_update + keys)
// -------------------------------------------------------------------
__global__ void k6_mem(const float* __restrict__ ws_qu,
                       const float* __restrict__ keys,
                       float* __restrict__ out_mem) {
  __shared__ float part[8];
  int t = threadIdx.x, m = blockIdx.x;
  float u = ws_qu[m * KDIM + t] + keys[m * KDIM + t];
  float p = u * u;
#pragma unroll
  for (int off = 16; off > 0; off >>= 1) p += __shfl_xor(p, off);
  if ((t & 31) == 0) part[t >> 5] = p;
  __syncthreads();
  float norm = 0.0f;
#pragma unroll
  for (int i = 0; i < 8; ++i) norm += part[i];
  norm = sqrtf(norm);
  out_mem[m * KDIM + t] = u / fmaxf(norm, 1e-12f);
}

// -------------------------------------------------------------------
extern "C" void kernel_launch(void* const* d_in, const int* in_sizes, int n_in,
                              void* d_out, int out_size, void* d_ws, size_t ws_size,
                              hipStream_t stream) {
  const float* q    = (const float*)d_in[0];   // (256,1024,64)
  const float* keys = (const float*)d_in[1];   // (64,256)

  float* out = (float*)d_out;
  float* out_uq     = out;                                 // 512*1024*64
  float* out_mem    = out_uq + (size_t)512 * NROW;         // 64*256
  float* out_sq     = out_mem + MKEYS * KDIM;              // 65536*64
  float* out_smem   = out_sq + (size_t)NROW * MKEYS;       // 65536*64
  float* out_spread = out_smem + (size_t)NROW * MKEYS;     // 65536
  float* out_gath   = out_spread + NROW;                   // 65536*256

  float*    ws_score  = (float*)d_ws;                      // NROW*64 f32
  unsigned* ws_colmax = (unsigned*)(ws_score + (size_t)NROW * MKEYS);
  float*    ws_colsum = (float*)(ws_colmax + MKEYS);
  int*      ws_g      = (int*)(ws_colsum + MKEYS);         // NROW
  float*    ws_w      = (float*)(ws_g + NROW);             // NROW
  float*    ws_qu     = ws_w + NROW;                       // 64*256

  k0_init   <<<1, 256, 0, stream>>>(ws_qu, ws_colmax, ws_colsum);
  k1_score  <<<NROW / ROWS1, 256, 0, stream>>>(q, keys, ws_score);
  k2_colmax <<<NROW / 256, 256, 0, stream>>>(ws_score, ws_colmax);
  k3_colsum <<<NROW / 256, 256, 0, stream>>>(ws_score, ws_colmax, ws_colsum);
  k4_rows   <<<NROW / TILE4, 256, 0, stream>>>(q, keys, ws_score, ws_colmax, ws_colsum,
                                               ws_g, ws_w, out_uq, out_sq, out_smem,
                                               out_spread, out_gath);
  k5_scatter<<<NROW / 1024, 256, 0, stream>>>(q, ws_g, ws_w, ws_qu);
  k6_mem    <<<MKEYS, 256, 0, stream>>>(ws_qu, keys, out_mem);
}